// GIN3_51728586113686
// MI455X (gfx1250) — compile-verified
//
#include <hip/hip_runtime.h>
#include <cstdint>

typedef __attribute__((ext_vector_type(2))) float v2f;
typedef __attribute__((ext_vector_type(8))) float v8f;

static __device__ __forceinline__ v8f relu8(v8f x) {
  v8f r;
#pragma unroll
  for (int i = 0; i < 8; ++i) r[i] = fmaxf(x[i], 0.0f);
  return r;
}

// float atomic max via monotone bit tricks (works for any sign mix)
static __device__ __forceinline__ void atomicMaxF(float* addr, float val) {
  if (val >= 0.0f) atomicMax((int*)addr, __float_as_int(val));
  else             atomicMin((unsigned int*)addr, __float_as_uint(val));
}

// ---------------- init kernels ----------------
__global__ void k_fill1(float* __restrict__ a, int n) {
  int i = blockIdx.x * blockDim.x + threadIdx.x;
  if (i < n) a[i] = 1.0f;
}

__global__ void k_init_scalars(float* __restrict__ s) {
  s[0] = -3.4e38f;  // running max of policy logits
  s[1] = 0.0f;      // sum v
  s[2] = 0.0f;      // sum v^2
  s[3] = 0.0f;      // sum exp
}

// ---------------- layer-0 aggregation: agg0 = 1 + degree ----------------
__global__ void k_degree(const int* __restrict__ row, float* __restrict__ agg0, int e) {
  int i = blockIdx.x * blockDim.x + threadIdx.x;
  if (i < e) atomicAdd(&agg0[row[i]], 1.0f);
}

// ---------------- layer-0 MLP (scalar input -> 8 features) ----------------
__global__ void k_mlp0(const float* __restrict__ agg0,
                       const float* __restrict__ W0, const float* __restrict__ b0,
                       const float* __restrict__ V0, const float* __restrict__ c0,
                       float* __restrict__ xout, int n) {
  int i = blockIdx.x * blockDim.x + threadIdx.x;
  if (i >= n) return;
  float a = agg0[i];
  float h[8], o[8];
#pragma unroll
  for (int j = 0; j < 8; ++j) h[j] = fmaxf(fmaf(a, W0[j], b0[j]), 0.0f);
#pragma unroll
  for (int f = 0; f < 8; ++f) o[f] = c0[f];
#pragma unroll
  for (int j = 0; j < 8; ++j)
#pragma unroll
    for (int f = 0; f < 8; ++f) o[f] = fmaf(h[j], V0[j * 8 + f], o[f]);
  float4 lo = make_float4(fmaxf(o[0],0.f), fmaxf(o[1],0.f), fmaxf(o[2],0.f), fmaxf(o[3],0.f));
  float4 hi = make_float4(fmaxf(o[4],0.f), fmaxf(o[5],0.f), fmaxf(o[6],0.f), fmaxf(o[7],0.f));
  ((float4*)xout)[(size_t)i * 2 + 0] = lo;
  ((float4*)xout)[(size_t)i * 2 + 1] = hi;
}

// ---------------- copy (self-loop term: agg = x) ----------------
__global__ void k_copy4(const float4* __restrict__ src, float4* __restrict__ dst, int n4) {
  int i = blockIdx.x * blockDim.x + threadIdx.x;
  if (i < n4) dst[i] = src[i];
}

// ---------------- edge scatter: agg[row] += x[col] (8 lanes of f32 atomics) -----
__global__ void k_scatter(const int* __restrict__ row, const int* __restrict__ col,
                          const float* __restrict__ x, float* __restrict__ agg, int e) {
  int i = blockIdx.x * blockDim.x + threadIdx.x;
  if (i >= e) return;
  int c = col[i];
  int r = row[i];
  const float4* xr = (const float4*)(x + (size_t)c * 8);
  float4 lo = xr[0], hi = xr[1];
  float* ar = agg + (size_t)r * 8;
  atomicAdd(ar + 0, lo.x); atomicAdd(ar + 1, lo.y);
  atomicAdd(ar + 2, lo.z); atomicAdd(ar + 3, lo.w);
  atomicAdd(ar + 4, hi.x); atomicAdd(ar + 5, hi.y);
  atomicAdd(ar + 6, hi.z); atomicAdd(ar + 7, hi.w);
}

// ---------------- WMMA MLP: xout = relu(relu(agg@W + b)@V + c), F=8 -----------
// One wave per 16-node tile. Two chained V_WMMA_F32_16X16X4_F32 per matmul
// (K=8 split 4+4), bias preloaded into C, exact fp32 arithmetic.
// Weight/bias loads use clamped indices + 0/1 mask instead of per-lane
// branches so no exec-mask churn is generated around the WMMA core.
__global__ void k_mlp_wmma(const float* __restrict__ agg,
                           const float* __restrict__ W, const float* __restrict__ b,
                           const float* __restrict__ V, const float* __restrict__ c,
                           float* __restrict__ xout, int n) {
  // per-wave staging: 16 rows x stride-10 floats (pad kills 2-way bank conflicts,
  // keeps 8B alignment for ds_load_b64)
  __shared__ float hbuf[8][160];
  const int lane = threadIdx.x & 31;
  const int wv   = threadIdx.x >> 5;
  const int tile = blockIdx.x * 8 + wv;
  const int base = tile * 16;

  const int m     = lane & 15;              // row index (A layout)
  const int khalf = (lane < 16) ? 0 : 2;    // A: lanes 16-31 carry K+2
  const int nn    = lane & 15;              // col index (B & C layouts)
  const int nc    = nn & 7;                 // clamped col (always-valid address)
  const float msk = (nn < 8) ? 1.0f : 0.0f; // zero the padded columns

  int node = base + m;
  if (node >= n) node = n - 1;              // clamp loads; keeps EXEC all-ones

  // A operand (16x8 agg tile), chunk0 = K0..3, chunk1 = K4..7
  v2f a0 = *(const v2f*)(agg + (size_t)node * 8 + 0 + khalf);
  v2f a1 = *(const v2f*)(agg + (size_t)node * 8 + 4 + khalf);

  // B operand from W (row-major 8x8), zero-padded to 16 columns (mask-multiply)
  v2f b0v, b1v;
  b0v.x = W[(0 + khalf) * 8 + nc] * msk;
  b0v.y = W[(1 + khalf) * 8 + nc] * msk;
  b1v.x = W[(4 + khalf) * 8 + nc] * msk;
  b1v.y = W[(5 + khalf) * 8 + nc] * msk;

  float bias = b[nc] * msk;
  v8f acc;
#pragma unroll
  for (int i = 0; i < 8; ++i) acc[i] = bias;

  acc = __builtin_amdgcn_wmma_f32_16x16x4_f32(false, a0, false, b0v, (short)0, acc, false, false);
  acc = __builtin_amdgcn_wmma_f32_16x16x4_f32(false, a1, false, b1v, (short)0, acc, false, false);
  acc = relu8(acc);

  // C-layout -> LDS row-major (transpose for next A operand)
  float* hb = &hbuf[wv][0];
  if (nn < 8) {
    const int rbase = (lane < 16) ? 0 : 8;  // C layout: vgpr r -> M=r (+8 upper half)
#pragma unroll
    for (int r = 0; r < 8; ++r) hb[(rbase + r) * 10 + nn] = acc[r];
  }
  __syncthreads();

  // second matmul: A' = h from LDS
  v2f a0b = *(const v2f*)(hb + m * 10 + 0 + khalf);
  v2f a1b = *(const v2f*)(hb + m * 10 + 4 + khalf);

  v2f b0w, b1w;
  b0w.x = V[(0 + khalf) * 8 + nc] * msk;
  b0w.y = V[(1 + khalf) * 8 + nc] * msk;
  b1w.x = V[(4 + khalf) * 8 + nc] * msk;
  b1w.y = V[(5 + khalf) * 8 + nc] * msk;

  float bias2 = c[nc] * msk;
  v8f acc2;
#pragma unroll
  for (int i = 0; i < 8; ++i) acc2[i] = bias2;

  acc2 = __builtin_amdgcn_wmma_f32_16x16x4_f32(false, a0b, false, b0w, (short)0, acc2, false, false);
  acc2 = __builtin_amdgcn_wmma_f32_16x16x4_f32(false, a1b, false, b1w, (short)0, acc2, false, false);
  acc2 = relu8(acc2);

  __syncthreads();
  if (nn < 8) {
    const int rbase = (lane < 16) ? 0 : 8;
#pragma unroll
    for (int r = 0; r < 8; ++r) hb[(rbase + r) * 10 + nn] = acc2[r];
  }
  __syncthreads();

  // coalesced b128 stores: lane<16 -> low half of row `lane`, lane>=16 -> high half
  {
    int r = lane & 15;
    int half = lane >> 4;
    int onode = base + r;
    if (onode < n) {
      const float* src = hb + r * 10 + half * 4;
      float4 v4 = make_float4(src[0], src[1], src[2], src[3]);
      *(float4*)(xout + (size_t)onode * 8 + half * 4) = v4;
    }
  }
}

// ---------------- heads: p_raw, v_raw + global max / sum / sum^2 --------------
__global__ void k_heads(const float* __restrict__ agg,
                        const float* __restrict__ Wp, const float* __restrict__ bp,
                        const float* __restrict__ Vp, const float* __restrict__ cp,
                        const float* __restrict__ Wv, const float* __restrict__ bv,
                        const float* __restrict__ Vv, const float* __restrict__ cv,
                        float* __restrict__ p_raw, float* __restrict__ v_raw,
                        float* __restrict__ scal, int n) {
  __shared__ float smax[256], ssum[256], ssum2[256];
  int i = blockIdx.x * blockDim.x + threadIdx.x;
  float p = -3.4e38f, v = 0.0f, v2 = 0.0f;
  if (i < n) {
    const float* a = agg + (size_t)i * 8;
    float dp = 0.0f, dv = 0.0f;
#pragma unroll
    for (int j = 0; j < 8; ++j) {
      dp = fmaf(a[j], Wp[j], dp);
      dv = fmaf(a[j], Wv[j], dv);
    }
    float hp = fmaxf(dp + bp[0], 0.0f);
    float hv = fmaxf(dv + bv[0], 0.0f);
    p = fmaf(hp, Vp[0], cp[0]);
    v = fmaf(hv, Vv[0], cv[0]);
    p_raw[i] = p;
    v_raw[i] = v;
    v2 = v * v;
  }
  smax[threadIdx.x] = p; ssum[threadIdx.x] = v; ssum2[threadIdx.x] = v2;
  __syncthreads();
  for (int s = 128; s > 0; s >>= 1) {
    if (threadIdx.x < s) {
      smax[threadIdx.x]  = fmaxf(smax[threadIdx.x], smax[threadIdx.x + s]);
      ssum[threadIdx.x] += ssum[threadIdx.x + s];
      ssum2[threadIdx.x] += ssum2[threadIdx.x + s];
    }
    __syncthreads();
  }
  if (threadIdx.x == 0) {
    atomicMaxF(&scal[0], smax[0]);
    atomicAdd(&scal[1], ssum[0]);
    atomicAdd(&scal[2], ssum2[0]);
  }
}

// ---------------- softmax pass 1: e = exp(p - max), accumulate sum -------------
__global__ void k_exp(const float* __restrict__ p_raw, const float* __restrict__ scal,
                      float* __restrict__ e_out, float* __restrict__ scal_w, int n) {
  __shared__ float ssum[256];
  int i = blockIdx.x * blockDim.x + threadIdx.x;
  float pmax = scal[0];
  float e = 0.0f;
  if (i < n) {
    e = expf(p_raw[i] - pmax);
    e_out[i] = e;
  }
  ssum[threadIdx.x] = e;
  __syncthreads();
  for (int s = 128; s > 0; s >>= 1) {
    if (threadIdx.x < s) ssum[threadIdx.x] += ssum[threadIdx.x + s];
    __syncthreads();
  }
  if (threadIdx.x == 0) atomicAdd(&scal_w[3], ssum[0]);
}

// ---------------- finalize: policy normalize + value standardize ---------------
__global__ void k_final(const float* __restrict__ v_raw, const float* __restrict__ scal,
                        float* __restrict__ out_policy, float* __restrict__ out_value,
                        int n) {
  int i = blockIdx.x * blockDim.x + threadIdx.x;
  if (i >= n) return;
  float inv_n = 1.0f / (float)n;
  float mean = scal[1] * inv_n;
  float var  = fmaf(scal[2], inv_n, -mean * mean);
  float inv_std = rsqrtf(var + 1e-10f);
  float inv_sum = 1.0f / scal[3];
  out_policy[i] = out_policy[i] * inv_sum;
  out_value[i]  = (v_raw[i] - mean) * inv_std;
}

extern "C" void kernel_launch(void* const* d_in, const int* in_sizes, int n_in,
                              void* d_out, int out_size, void* d_ws, size_t ws_size,
                              hipStream_t stream) {
  const int* row = (const int*)d_in[0];
  const int* col = (const int*)d_in[1];
  const int E = in_sizes[0];
  const int N = out_size / 2;   // output = policy(N) ++ value(N)

  const float* W0 = (const float*)d_in[3];
  const float* b0 = (const float*)d_in[4];
  const float* V0 = (const float*)d_in[5];
  const float* c0 = (const float*)d_in[6];
  const float* W1 = (const float*)d_in[7];
  const float* b1 = (const float*)d_in[8];
  const float* V1 = (const float*)d_in[9];
  const float* c1 = (const float*)d_in[10];
  const float* Wp = (const float*)d_in[11];
  const float* bp = (const float*)d_in[12];
  const float* Vp = (const float*)d_in[13];
  const float* cp = (const float*)d_in[14];
  const float* Wv = (const float*)d_in[15];
  const float* bv = (const float*)d_in[16];
  const float* Vv = (const float*)d_in[17];
  const float* cv = (const float*)d_in[18];

  char* ws = (char*)d_ws;
  float* bufX  = (float*)ws;                          // N*8 floats
  float* bufA  = (float*)(ws + (size_t)N * 8 * 4);    // N*8 floats
  float* agg0  = (float*)(ws + (size_t)N * 16 * 4);   // N floats
  float* p_raw = (float*)(ws + (size_t)N * 17 * 4);   // N floats
  float* v_raw = (float*)(ws + (size_t)N * 18 * 4);   // N floats
  float* scal  = (float*)(ws + (size_t)N * 19 * 4);   // 4 floats

  float* out_policy = (float*)d_out;
  float* out_value  = out_policy + N;

  const int B = 256;
  const int gN  = (N + B - 1) / B;
  const int gE  = (E + B - 1) / B;
  const int gN4 = (N * 2 + B - 1) / B;       // N*2 float4s
  const int tiles = (N + 15) / 16;
  const int gT  = (tiles + 7) / 8;           // 8 waves (tiles) per block

  k_init_scalars<<<1, 1, 0, stream>>>(scal);

  // Layer 0: agg0 = 1 + degree (x is all-ones, so gather collapses to counting)
  k_fill1 <<<gN, B, 0, stream>>>(agg0, N);
  k_degree<<<gE, B, 0, stream>>>(row, agg0, E);
  k_mlp0  <<<gN, B, 0, stream>>>(agg0, W0, b0, V0, c0, bufX, N);

  // Layer 1: agg = x + scatter(x); MLP via fp32 WMMA
  k_copy4   <<<gN4, B, 0, stream>>>((const float4*)bufX, (float4*)bufA, N * 2);
  k_scatter <<<gE,  B, 0, stream>>>(row, col, bufX, bufA, E);
  k_mlp_wmma<<<gT,  B, 0, stream>>>(bufA, W1, b1, V1, c1, bufX, N);

  // Final aggregation for heads
  k_copy4   <<<gN4, B, 0, stream>>>((const float4*)bufX, (float4*)bufA, N * 2);
  k_scatter <<<gE,  B, 0, stream>>>(row, col, bufX, bufA, E);

  // Heads + global reductions + softmax + standardization
  k_heads<<<gN, B, 0, stream>>>(bufA, Wp, bp, Vp, cp, Wv, bv, Vv, cv,
                                p_raw, v_raw, scal, N);
  k_exp  <<<gN, B, 0, stream>>>(p_raw, scal, out_policy, scal, N);
  k_final<<<gN, B, 0, stream>>>(v_raw, scal, out_policy, out_value, N);
}